// GNN_flexible_20358144983396
// MI455X (gfx1250) — compile-verified
//
#include <hip/hip_runtime.h>

#define N_NODES   100000
#define N_EDGES   600000
#define NUM_GRAPHS 256

typedef __attribute__((ext_vector_type(2))) float v2f;
typedef __attribute__((ext_vector_type(8))) float v8f;

// ---------------------------------------------------------------- zero
__global__ void zero_f32(float* __restrict__ p, int n) {
  int i = blockIdx.x * blockDim.x + threadIdx.x;
  if (i < n) p[i] = 0.0f;
}

// ---------------------------------------------------------------- edge gather + scatter-add
// grid = (ceil(E/256), Fin); block = 256
__global__ __launch_bounds__(256) void edge_scatter(
    const float* __restrict__ x, const int* __restrict__ src,
    const int* __restrict__ dst, const float* __restrict__ w,
    float* __restrict__ agg, int E, int Fin) {
  int e = blockIdx.x * blockDim.x + threadIdx.x;
  int f = blockIdx.y;
  if (e < E) {
    int s = src[e];
    int d = dst[e];
    float v = x[s * Fin + f] * w[e];
    atomicAdd(&agg[d * Fin + f], v);  // global_atomic_add_f32, L2-resident
  }
}

// ---------------------------------------------------------------- fused node update (WMMA fp32)
// y = relu(agg @ Wrel + brel + xin @ Wroot)
// block = 256 threads = 8 waves; each wave owns a 16-node row stripe and
// register-blocks ALL FOUT/16 n-tiles, so A fragments are loaded once per
// K-step and reused across the full output width.
// grid = ceil(nMtiles/8)
template <int FIN, int FOUT>
__global__ __launch_bounds__(256) void node_update(
    const float* __restrict__ agg, const float* __restrict__ xin,
    const float* __restrict__ Wrel, const float* __restrict__ Wroot,
    const float* __restrict__ brel, float* __restrict__ y, int nMtiles) {
  constexpr int NT = FOUT / 16;  // n-tiles register-blocked per wave

  const int wave = threadIdx.x >> 5;  // wave32
  const int lane = threadIdx.x & 31;
  const int half = lane >> 4;         // 0: lanes 0-15, 1: lanes 16-31
  const int lm   = lane & 15;

  const int mtile = blockIdx.x * 8 + wave;
  if (mtile >= nMtiles) return;       // wave-uniform; WMMAs run with EXEC all-1s
  const int m0   = mtile * 16;
  const int rowA = m0 + lm;           // A fragment: row = lane%16, k = 2*half + vgpr

  v8f acc[NT] = {};                   // independent chains -> no WMMA RAW hazard
  constexpr int KSTEPS = (FIN + 3) / 4;
#pragma unroll 4
  for (int kk = 0; kk < KSTEPS; ++kk) {
    const int k0 = kk * 4 + 2 * half;
    v2f a_agg, a_x;
    if constexpr ((FIN & 3) == 0) {
      a_agg.x = agg[rowA * FIN + k0];
      a_agg.y = agg[rowA * FIN + k0 + 1];
      a_x.x   = xin[rowA * FIN + k0];
      a_x.y   = xin[rowA * FIN + k0 + 1];
    } else {  // FIN==5: zero-pad K to a multiple of 4
      const bool p0 = (k0 < FIN), p1 = (k0 + 1 < FIN);
      a_agg.x = p0 ? agg[rowA * FIN + k0] : 0.0f;
      a_agg.y = p1 ? agg[rowA * FIN + k0 + 1] : 0.0f;
      a_x.x   = p0 ? xin[rowA * FIN + k0] : 0.0f;
      a_x.y   = p1 ? xin[rowA * FIN + k0 + 1] : 0.0f;
    }
#pragma unroll
    for (int nt = 0; nt < NT; ++nt) {
      const int n0 = nt * 16;
      v2f b_rel, b_root;  // weights are <=64KB total: hot in WGP$
      if constexpr ((FIN & 3) == 0) {
        b_rel.x  = Wrel[k0 * FOUT + n0 + lm];
        b_rel.y  = Wrel[(k0 + 1) * FOUT + n0 + lm];
        b_root.x = Wroot[k0 * FOUT + n0 + lm];
        b_root.y = Wroot[(k0 + 1) * FOUT + n0 + lm];
      } else {
        const bool p0 = (k0 < FIN), p1 = (k0 + 1 < FIN);
        b_rel.x  = p0 ? Wrel[k0 * FOUT + n0 + lm] : 0.0f;
        b_rel.y  = p1 ? Wrel[(k0 + 1) * FOUT + n0 + lm] : 0.0f;
        b_root.x = p0 ? Wroot[k0 * FOUT + n0 + lm] : 0.0f;
        b_root.y = p1 ? Wroot[(k0 + 1) * FOUT + n0 + lm] : 0.0f;
      }
      // D = A*B + C, fp32 in / fp32 accumulate (matches fp32 reference)
      acc[nt] = __builtin_amdgcn_wmma_f32_16x16x4_f32(
          false, a_agg, false, b_rel, (short)0, acc[nt], false, false);
      acc[nt] = __builtin_amdgcn_wmma_f32_16x16x4_f32(
          false, a_x, false, b_root, (short)0, acc[nt], false, false);
    }
  }

  // epilogue: bias + ReLU; C/D layout: VGPR v, lane -> (M = v + 8*half, N = lane%16)
#pragma unroll
  for (int nt = 0; nt < NT; ++nt) {
    const int n0 = nt * 16;
    const float bias = brel[n0 + lm];
#pragma unroll
    for (int v = 0; v < 8; ++v) {
      const int row = m0 + v + 8 * half;
      float val = acc[nt][v] + bias;
      y[row * FOUT + n0 + lm] = val > 0.0f ? val : 0.0f;
    }
  }
}

// ---------------------------------------------------------------- global mean pool (scatter)
// thread t -> node = t/32, feature = t%32
__global__ __launch_bounds__(256) void pool_scatter(
    const float* __restrict__ x, const int* __restrict__ batch,
    float* __restrict__ sums, float* __restrict__ cnts, int N) {
  int t = blockIdx.x * blockDim.x + threadIdx.x;
  int node = t >> 5;
  int f = t & 31;
  if (node < N) {
    int g = batch[node];
    atomicAdd(&sums[g * 32 + f], x[node * 32 + f]);
    if (f == 0) atomicAdd(&cnts[g], 1.0f);
  }
}

// ---------------------------------------------------------------- mean + 3-layer MLP (tiny)
__global__ __launch_bounds__(256) void pool_mlp(
    const float* __restrict__ sums, const float* __restrict__ cnts,
    const float* __restrict__ W0, const float* __restrict__ b0,
    const float* __restrict__ W1, const float* __restrict__ b1,
    const float* __restrict__ W2, const float* __restrict__ b2,
    float* __restrict__ out) {
  int g = threadIdx.x;  // one graph per thread, 256 threads
  float inv = 1.0f / fmaxf(cnts[g], 1.0f);
  float h0[32], h1[32], h2[16];
#pragma unroll
  for (int i = 0; i < 32; ++i) h0[i] = sums[g * 32 + i] * inv;
  for (int j = 0; j < 32; ++j) {
    float s = b0[j];
    for (int i = 0; i < 32; ++i) s = fmaf(h0[i], W0[i * 32 + j], s);
    h1[j] = fmaxf(s, 0.0f);
  }
  for (int j = 0; j < 16; ++j) {
    float s = b1[j];
    for (int i = 0; i < 32; ++i) s = fmaf(h1[i], W1[i * 16 + j], s);
    h2[j] = fmaxf(s, 0.0f);
  }
  float s = b2[0];
  for (int i = 0; i < 16; ++i) s = fmaf(h2[i], W2[i], s);
  out[g] = s;
}

// ---------------------------------------------------------------- per-layer driver
template <int FIN, int FOUT>
static void run_layer(const float* xin, const int* src, const int* dst,
                      const float* ew, const float* Wrel, const float* brel,
                      const float* Wroot, float* agg, float* y, hipStream_t s) {
  int n = N_NODES * FIN;
  zero_f32<<<(n + 255) / 256, 256, 0, s>>>(agg, n);
  edge_scatter<<<dim3((N_EDGES + 255) / 256, FIN), 256, 0, s>>>(
      xin, src, dst, ew, agg, N_EDGES, FIN);
  const int nMtiles = N_NODES / 16;  // 6250 exactly
  node_update<FIN, FOUT><<<(nMtiles + 7) / 8, 256, 0, s>>>(
      agg, xin, Wrel, Wroot, brel, y, nMtiles);
}

extern "C" void kernel_launch(void* const* d_in, const int* in_sizes, int n_in,
                              void* d_out, int out_size, void* d_ws, size_t ws_size,
                              hipStream_t stream) {
  const float* x0    = (const float*)d_in[0];
  const int*   ei    = (const int*)d_in[1];
  const float* ew    = (const float*)d_in[2];
  const int*   batch = (const int*)d_in[3];
  const int* src = ei;
  const int* dst = ei + N_EDGES;

  // GraphConv params: (W_rel, b_rel, W_root) x 5 starting at index 4
  const float* Wr[5];
  const float* br[5];
  const float* Ws[5];
  for (int i = 0; i < 5; ++i) {
    Wr[i] = (const float*)d_in[4 + 3 * i];
    br[i] = (const float*)d_in[5 + 3 * i];
    Ws[i] = (const float*)d_in[6 + 3 * i];
  }
  const float* mW0 = (const float*)d_in[19];
  const float* mb0 = (const float*)d_in[20];
  const float* mW1 = (const float*)d_in[21];
  const float* mb1 = (const float*)d_in[22];
  const float* mW2 = (const float*)d_in[23];
  const float* mb2 = (const float*)d_in[24];

  // workspace layout (floats): xA[N*128] | xB[N*64] | agg[N*128] | sums[256*32] | cnts[256]
  float* ws   = (float*)d_ws;
  float* xA   = ws;
  float* xB   = xA + (size_t)N_NODES * 128;
  float* agg  = xB + (size_t)N_NODES * 64;
  float* sums = agg + (size_t)N_NODES * 128;
  float* cnts = sums + NUM_GRAPHS * 32;

  // 5 GraphConv layers: 5->32->64->128->64->32 (ping-pong xA/xB)
  run_layer<5, 32>(x0, src, dst, ew, Wr[0], br[0], Ws[0], agg, xA, stream);
  run_layer<32, 64>(xA, src, dst, ew, Wr[1], br[1], Ws[1], agg, xB, stream);
  run_layer<64, 128>(xB, src, dst, ew, Wr[2], br[2], Ws[2], agg, xA, stream);
  run_layer<128, 64>(xA, src, dst, ew, Wr[3], br[3], Ws[3], agg, xB, stream);
  run_layer<64, 32>(xB, src, dst, ew, Wr[4], br[4], Ws[4], agg, xA, stream);

  // global mean pool + MLP head
  zero_f32<<<(NUM_GRAPHS * 32 + NUM_GRAPHS + 255) / 256, 256, 0, stream>>>(
      sums, NUM_GRAPHS * 32 + NUM_GRAPHS);
  pool_scatter<<<(N_NODES * 32 + 255) / 256, 256, 0, stream>>>(xA, batch, sums,
                                                               cnts, N_NODES);
  pool_mlp<<<1, 256, 0, stream>>>(sums, cnts, mW0, mb0, mW1, mb1, mW2, mb2,
                                  (float*)d_out);
}